// GCNet_36627481101155
// MI455X (gfx1250) — compile-verified
//
#include <hip/hip_runtime.h>
#include <math.h>

#define N_NODES 50000
#define N_EDGES 800000
#define IN_CH   128
#define HID_CH  256
#define OUT_CH  40
#define M_TILES (N_NODES / 16)   // 3125, exact

typedef __attribute__((ext_vector_type(2))) float v2f;
typedef __attribute__((ext_vector_type(8))) float v8f;

// ---------------- degree / norm ----------------
__global__ void k_fill1(float* __restrict__ p, int n) {
    int i = blockIdx.x * blockDim.x + threadIdx.x;
    if (i < n) p[i] = 1.0f;                       // self-loop contributes 1 to deg
}

__global__ void k_count_deg(const int* __restrict__ dst, float* __restrict__ deg) {
    int e = blockIdx.x * blockDim.x + threadIdx.x;
    if (e < N_EDGES) atomicAdd(&deg[dst[e]], 1.0f);
}

__global__ void k_rsqrt(float* __restrict__ dinv, int n) {
    int i = blockIdx.x * blockDim.x + threadIdx.x;
    if (i < n) dinv[i] = rsqrtf(dinv[i]);         // deg >= 1 always (self-loop)
}

// ---------------- GEMM1: X[50000,128] @ W1[128,256] via V_WMMA_F32_16X16X4_F32 ----------------
// One wave computes a 16x64 output strip (4 N-tiles), reusing the A fragment.
__global__ void k_gemm1(const float* __restrict__ X, const float* __restrict__ W,
                        float* __restrict__ O) {
    const int wave = (int)((blockIdx.x * blockDim.x + threadIdx.x) >> 5);
    const int lane = threadIdx.x & 31;
    const int NG = HID_CH / 64;                   // 4 groups of 64 columns
    const int mt = wave / NG;
    const int ng = wave % NG;
    if (mt >= M_TILES) return;                    // whole-wave uniform exit (EXEC stays all-1s)
    const int m0   = mt * 16;
    const int n0   = ng * 64;
    const int half = lane >> 4;                   // 0: K=0,1   1: K=2,3
    const int l15  = lane & 15;

    const float* xrow = X + (size_t)(m0 + l15) * IN_CH + 2 * half;
    v8f c0 = {}, c1 = {}, c2 = {}, c3 = {};

    for (int k0 = 0; k0 < IN_CH; k0 += 4) {
        v2f a = *(const v2f*)(xrow + k0);         // A: rows m0+l15, K = k0+2*half .. +1
        const float* w0 = W + (size_t)(k0 + 2 * half) * HID_CH + n0 + l15;
        const float* w1 = w0 + HID_CH;
        v2f b0 = { w0[0],  w1[0]  };
        v2f b1 = { w0[16], w1[16] };
        v2f b2 = { w0[32], w1[32] };
        v2f b3 = { w0[48], w1[48] };
        c0 = __builtin_amdgcn_wmma_f32_16x16x4_f32(false, a, false, b0, (short)0, c0, false, false);
        c1 = __builtin_amdgcn_wmma_f32_16x16x4_f32(false, a, false, b1, (short)0, c1, false, false);
        c2 = __builtin_amdgcn_wmma_f32_16x16x4_f32(false, a, false, b2, (short)0, c2, false, false);
        c3 = __builtin_amdgcn_wmma_f32_16x16x4_f32(false, a, false, b3, (short)0, c3, false, false);
    }

    // C/D layout: VGPR r -> row m0 + r + 8*half, col n0 + l15 (+0/16/32/48)
    float* orow = O + (size_t)(m0 + 8 * half) * HID_CH + n0 + l15;
#pragma unroll
    for (int r = 0; r < 8; ++r) {
        orow[(size_t)r * HID_CH + 0]  = c0[r];
        orow[(size_t)r * HID_CH + 16] = c1[r];
        orow[(size_t)r * HID_CH + 32] = c2[r];
        orow[(size_t)r * HID_CH + 48] = c3[r];
    }
}

// ---------------- GEMM2: H[50000,256] @ W2[256,40] (N padded to 48, guarded) ----------------
__global__ void k_gemm2(const float* __restrict__ H, const float* __restrict__ W,
                        float* __restrict__ O) {
    const int wave = (int)((blockIdx.x * blockDim.x + threadIdx.x) >> 5);
    const int lane = threadIdx.x & 31;
    if (wave >= M_TILES) return;
    const int m0   = wave * 16;
    const int half = lane >> 4;
    const int l15  = lane & 15;

    const float* hrow = H + (size_t)(m0 + l15) * HID_CH + 2 * half;
    v8f c0 = {}, c1 = {}, c2 = {};
    const bool in2 = (32 + l15) < OUT_CH;         // tile-2 columns 32..47: valid only <40

    for (int k0 = 0; k0 < HID_CH; k0 += 4) {
        v2f a = *(const v2f*)(hrow + k0);
        const float* w0 = W + (size_t)(k0 + 2 * half) * OUT_CH + l15;
        const float* w1 = w0 + OUT_CH;
        v2f b0 = { w0[0],  w1[0]  };
        v2f b1 = { w0[16], w1[16] };
        v2f b2;
        b2.x = in2 ? w0[32] : 0.0f;
        b2.y = in2 ? w1[32] : 0.0f;
        c0 = __builtin_amdgcn_wmma_f32_16x16x4_f32(false, a, false, b0, (short)0, c0, false, false);
        c1 = __builtin_amdgcn_wmma_f32_16x16x4_f32(false, a, false, b1, (short)0, c1, false, false);
        c2 = __builtin_amdgcn_wmma_f32_16x16x4_f32(false, a, false, b2, (short)0, c2, false, false);
    }

    float* orow = O + (size_t)(m0 + 8 * half) * OUT_CH + l15;   // stride OUT_CH = 40
#pragma unroll
    for (int r = 0; r < 8; ++r) {
        orow[(size_t)r * OUT_CH + 0]  = c0[r];
        orow[(size_t)r * OUT_CH + 16] = c1[r];
        if (in2) orow[(size_t)r * OUT_CH + 32] = c2[r];
    }
}

// ---------------- layer-1 aggregation ----------------
// init: agg[i] = XW[i] * dinv[i]^2   (self-loop term; fully overwrites poisoned ws)
__global__ void k_selfinit256(const float* __restrict__ xw, const float* __restrict__ dinv,
                              float* __restrict__ agg) {
    int idx = blockIdx.x * blockDim.x + threadIdx.x;     // N*64 float4 groups
    if (idx >= N_NODES * 64) return;
    int node = idx >> 6;
    int g = (idx & 63) * 4;
    float di = dinv[node];
    float s = di * di;
    const float4 v = *(const float4*)(xw + (size_t)node * HID_CH + g);
    float4 r; r.x = v.x * s; r.y = v.y * s; r.z = v.z * s; r.w = v.w * s;
    *(float4*)(agg + (size_t)node * HID_CH + g) = r;
}

__global__ void k_edge256(const float* __restrict__ xw, const int* __restrict__ src,
                          const int* __restrict__ dst, const float* __restrict__ dinv,
                          float* __restrict__ agg) {
    long long idx = (long long)blockIdx.x * blockDim.x + threadIdx.x;  // E*64 groups
    int e = (int)(idx >> 6);
    if (e >= N_EDGES) return;
    int g = ((int)idx & 63) * 4;
    int s = src[e], d = dst[e];
    float nrm = dinv[s] * dinv[d];
    const float4 v = *(const float4*)(xw + (size_t)s * HID_CH + g);
    float* p = agg + (size_t)d * HID_CH + g;
    atomicAdd(p + 0, v.x * nrm);
    atomicAdd(p + 1, v.y * nrm);
    atomicAdd(p + 2, v.z * nrm);
    atomicAdd(p + 3, v.w * nrm);
}

__global__ void k_bias_relu(float* __restrict__ h, const float* __restrict__ b1) {
    int idx = blockIdx.x * blockDim.x + threadIdx.x;     // N*64 float4 groups
    if (idx >= N_NODES * 64) return;
    int node = idx >> 6;
    int g = (idx & 63) * 4;
    float4 v = *(const float4*)(h + (size_t)node * HID_CH + g);
    const float4 b = *(const float4*)(b1 + g);
    v.x = fmaxf(v.x + b.x, 0.0f);
    v.y = fmaxf(v.y + b.y, 0.0f);
    v.z = fmaxf(v.z + b.z, 0.0f);
    v.w = fmaxf(v.w + b.w, 0.0f);
    *(float4*)(h + (size_t)node * HID_CH + g) = v;
}

// ---------------- layer-2 aggregation (into d_out) ----------------
__global__ void k_selfinit40(const float* __restrict__ o1, const float* __restrict__ dinv,
                             float* __restrict__ out) {
    int idx = blockIdx.x * blockDim.x + threadIdx.x;     // N*40 elements
    if (idx >= N_NODES * OUT_CH) return;
    int node = idx / OUT_CH;
    float di = dinv[node];
    out[idx] = o1[idx] * di * di;
}

__global__ void k_edge40(const float* __restrict__ o1, const int* __restrict__ src,
                         const int* __restrict__ dst, const float* __restrict__ dinv,
                         float* __restrict__ out) {
    long long idx = (long long)blockIdx.x * blockDim.x + threadIdx.x;  // E*40
    if (idx >= (long long)N_EDGES * OUT_CH) return;
    int e = (int)(idx / OUT_CH);
    int c = (int)(idx - (long long)e * OUT_CH);
    int s = src[e], d = dst[e];
    float nrm = dinv[s] * dinv[d];
    atomicAdd(&out[(size_t)d * OUT_CH + c], o1[(size_t)s * OUT_CH + c] * nrm);
}

__global__ void k_logsoftmax(float* __restrict__ out, const float* __restrict__ b2) {
    int i = blockIdx.x * blockDim.x + threadIdx.x;
    if (i >= N_NODES) return;
    float* p = out + (size_t)i * OUT_CH;
    float v[OUT_CH];
    float m = -INFINITY;
#pragma unroll
    for (int c = 0; c < OUT_CH; ++c) { v[c] = p[c] + b2[c]; m = fmaxf(m, v[c]); }
    float ssum = 0.0f;
#pragma unroll
    for (int c = 0; c < OUT_CH; ++c) ssum += expf(v[c] - m);
    float l = m + logf(ssum);
#pragma unroll
    for (int c = 0; c < OUT_CH; ++c) p[c] = v[c] - l;
}

extern "C" void kernel_launch(void* const* d_in, const int* in_sizes, int n_in,
                              void* d_out, int out_size, void* d_ws, size_t ws_size,
                              hipStream_t stream) {
    const float* x  = (const float*)d_in[0];
    const int*   ei = (const int*)d_in[1];
    const float* w1 = (const float*)d_in[2];
    const float* b1 = (const float*)d_in[3];
    const float* w2 = (const float*)d_in[4];
    const float* b2 = (const float*)d_in[5];
    const int* src = ei;                 // edge_index row 0
    const int* dst = ei + N_EDGES;       // edge_index row 1
    float* out = (float*)d_out;

    float* xw   = (float*)d_ws;                                  // [N, 256]
    float* h    = xw + (size_t)N_NODES * HID_CH;                 // [N, 256]
    float* dinv = h  + (size_t)N_NODES * HID_CH;                 // [N]
    float* o1   = xw;                                            // reuse after layer 1

    // degree norm: dinv = rsqrt(indeg + 1)
    k_fill1    <<<(N_NODES + 255) / 256, 256, 0, stream>>>(dinv, N_NODES);
    k_count_deg<<<(N_EDGES + 255) / 256, 256, 0, stream>>>(dst, dinv);
    k_rsqrt    <<<(N_NODES + 255) / 256, 256, 0, stream>>>(dinv, N_NODES);

    // layer 1
    k_gemm1<<<(M_TILES * 4 + 7) / 8, 256, 0, stream>>>(x, w1, xw);             // 12500 waves
    k_selfinit256<<<(N_NODES * 64 + 255) / 256, 256, 0, stream>>>(xw, dinv, h);
    k_edge256<<<(int)(((long long)N_EDGES * 64 + 255) / 256), 256, 0, stream>>>(xw, src, dst, dinv, h);
    k_bias_relu<<<(N_NODES * 64 + 255) / 256, 256, 0, stream>>>(h, b1);

    // layer 2
    k_gemm2<<<(M_TILES + 7) / 8, 256, 0, stream>>>(h, w2, o1);                 // 3125 waves
    k_selfinit40<<<(N_NODES * OUT_CH + 255) / 256, 256, 0, stream>>>(o1, dinv, out);
    k_edge40<<<(int)(((long long)N_EDGES * OUT_CH + 255) / 256), 256, 0, stream>>>(o1, src, dst, dinv, out);
    k_logsoftmax<<<(N_NODES + 255) / 256, 256, 0, stream>>>(out, b2);
}